// MeshDownConv_89489938579755
// MI455X (gfx1250) — compile-verified
//
#include <hip/hip_runtime.h>
#include <hip/hip_bf16.h>

// Problem constants (from reference)
#define NB 4
#define NN 50000
#define NE 800000
#define NC 32
#define NO 32
#define NK 9            // 9 spline taps, flattened weight rows = 288
#define KTOT (NK * NC)  // 288
#define NPAIR (KTOT / 2)

typedef __attribute__((ext_vector_type(2))) float v2f;
typedef __attribute__((ext_vector_type(8))) float v8f;

// ---------------------------------------------------------------------------
// init: out[n,o] = bias[o] + sum_c act(x[n,c]) * root[c,o]
// ---------------------------------------------------------------------------
template <bool RELU>
__global__ __launch_bounds__(256) void spline_init_kernel(
    const float* __restrict__ x, const float* __restrict__ root,
    const float* __restrict__ bias, float* __restrict__ out) {
  int idx = blockIdx.x * 256 + threadIdx.x;
  if (idx >= NN * NO) return;
  int n = idx >> 5;
  int o = idx & 31;
  const float* xr = x + (long)n * NC;
  float acc = bias[o];
#pragma unroll
  for (int c = 0; c < NC; ++c) {
    float v = xr[c];
    if (RELU) v = fmaxf(v, 0.0f);
    acc += v * root[c * NO + o];
  }
  out[idx] = acc;
}

// ---------------------------------------------------------------------------
// relu, in place (final layer output only)
// ---------------------------------------------------------------------------
__global__ __launch_bounds__(256) void spline_relu_kernel(float* __restrict__ h) {
  int idx = blockIdx.x * 256 + threadIdx.x;
  if (idx < NN * NO) h[idx] = fmaxf(h[idx], 0.0f);
}

// ---------------------------------------------------------------------------
// edge kernel: per wave, 32 edges (two 16-row M-tiles), 4 WMMA acc chains.
//   A [32 x 288] : A[m, k*32+c] = basis[m,k] * act(x[src[m], c])
//   Bw [288 x 32]: spline weights, staged in LDS pre-swizzled into b64
//                  fragment pairs: wswz[p*64 + col*2 + q] = Wflat[2p+q][col]
//   D = A x Bw via V_WMMA_F32_16X16X4_F32, scatter-added with relaxed
//   agent-scope GLOBAL_ATOMIC_ADD_F32.
// 128 threads = 4 waves x 32 edges = 128 edges/block; E = 6250 * 128 exactly.
// One lane owns one edge: no divergence, single barrier.
// ---------------------------------------------------------------------------
template <bool RELU>
__global__ __launch_bounds__(128) void spline_edge_kernel(
    const float* __restrict__ x, const int* __restrict__ src,
    const int* __restrict__ dst, const float* __restrict__ ps,
    const float* __restrict__ Wk, float* __restrict__ out) {
  __shared__ float wswz[NPAIR * 64];    // 288x32 weights, fragment-swizzled: 36 KB
  __shared__ float xjs[4][32][NC];      // gathered features: 16 KB
  __shared__ float bas[4][32][NK];      // basis values: 4.5 KB
  __shared__ int   dsts[4][32];

  const int tid  = threadIdx.x;
  const int w    = tid >> 5;            // wave id within block (0..3)
  const int lane = tid & 31;
  const int lid  = lane & 15;
  const int half = lane >> 4;
  const long e0  = ((long)blockIdx.x * 4 + w) * 32;

  // cooperative: weights -> LDS, swizzled so a lane's B fragment is one b64
  for (int i = tid; i < KTOT * NO; i += 128) {
    int row = i >> 5, col = i & 31;
    wswz[(row >> 1) * 64 + col * 2 + (row & 1)] = Wk[i];
  }

  // each lane owns edge e0+lane: metadata, basis, feature-row gather
  {
    long e = e0 + lane;
    int  s = src[e];
    dsts[w][lane] = dst[e];
    float t0 = ps[2 * e];
    float t1 = ps[2 * e + 1];
    float u0 = 1.0f - t0, u1 = 1.0f - t1;
    float b0[3], b1[3];
    b0[0] = 0.5f * u0 * u0; b0[1] = -t0 * t0 + t0 + 0.5f; b0[2] = 0.5f * t0 * t0;
    b1[0] = 0.5f * u1 * u1; b1[1] = -t1 * t1 + t1 + 0.5f; b1[2] = 0.5f * t1 * t1;
#pragma unroll
    for (int k1 = 0; k1 < 3; ++k1)
#pragma unroll
      for (int k0 = 0; k0 < 3; ++k0)
        bas[w][lane][k1 * 3 + k0] = b1[k1] * b0[k0];   // k = k1*3 + k0 (k1 major)

    const float4* xr = (const float4*)(x + (long)s * NC);
#pragma unroll
    for (int i = 0; i < 8; ++i) {
      float4 v = xr[i];
      if (RELU) {
        v.x = fmaxf(v.x, 0.0f); v.y = fmaxf(v.y, 0.0f);
        v.z = fmaxf(v.z, 0.0f); v.w = fmaxf(v.w, 0.0f);
      }
      xjs[w][lane][4 * i + 0] = v.x;
      xjs[w][lane][4 * i + 1] = v.y;
      xjs[w][lane][4 * i + 2] = v.z;
      xjs[w][lane][4 * i + 3] = v.w;
    }
  }
  __syncthreads();

  // K loop: 9 taps x 8 steps of K=4; 2 M-tiles x 2 N-tiles -> 288 WMMAs / wave
  v8f acc[2][2];
#pragma unroll
  for (int mt = 0; mt < 2; ++mt)
#pragma unroll
    for (int j = 0; j < 2; ++j)
      acc[mt][j] = (v8f){0.f, 0.f, 0.f, 0.f, 0.f, 0.f, 0.f, 0.f};

  const float* xrow0 = &xjs[w][lid][0];
  const float* xrow1 = &xjs[w][lid + 16][0];
  for (int tap = 0; tap < NK; ++tap) {
    float bs0 = bas[w][lid][tap];
    float bs1 = bas[w][lid + 16][tap];
#pragma unroll
    for (int tt = 0; tt < 8; ++tt) {
      int cc = tt * 4 + 2 * half;                 // K-offset inside this tap
      int p  = tap * 16 + tt * 2 + half;          // fragment pair index
      // A fragments: lane holds row M=lid (tile 0) / lid+16 (tile 1), K={cc,cc+1}
      v2f a0, a1;
      a0.x = bs0 * xrow0[cc]; a0.y = bs0 * xrow0[cc + 1];
      a1.x = bs1 * xrow1[cc]; a1.y = bs1 * xrow1[cc + 1];
      // B fragments: one aligned b64 per N-tile, lands in adjacent VGPR pair
      const v2f* bp = (const v2f*)&wswz[p * 64];
      v2f b0 = bp[lid];
      v2f b1 = bp[lid + 16];
      acc[0][0] = __builtin_amdgcn_wmma_f32_16x16x4_f32(false, a0, false, b0,
                                                        (short)0, acc[0][0], false, false);
      acc[0][1] = __builtin_amdgcn_wmma_f32_16x16x4_f32(false, a0, false, b1,
                                                        (short)0, acc[0][1], false, false);
      acc[1][0] = __builtin_amdgcn_wmma_f32_16x16x4_f32(false, a1, false, b0,
                                                        (short)0, acc[1][0], false, false);
      acc[1][1] = __builtin_amdgcn_wmma_f32_16x16x4_f32(false, a1, false, b1,
                                                        (short)0, acc[1][1], false, false);
    }
  }

  // scatter-add: tile (mt,j) element r -> (M = 16*mt + 8*half + r, N = 16*j + lid)
#pragma unroll
  for (int mt = 0; mt < 2; ++mt) {
#pragma unroll
    for (int j = 0; j < 2; ++j) {
      int o = j * 16 + lid;
#pragma unroll
      for (int r = 0; r < 8; ++r) {
        int m = mt * 16 + half * 8 + r;
        int d = dsts[w][m];
        __hip_atomic_fetch_add(&out[(long)d * NO + o], acc[mt][j][r],
                               __ATOMIC_RELAXED, __HIP_MEMORY_SCOPE_AGENT);
      }
    }
  }
}

// ---------------------------------------------------------------------------
// launcher: per batch: init -> edge (layer 1, pre-ReLU h) ->
//           init/edge reading relu(h) on the fly (layer 2) -> final relu.
// ---------------------------------------------------------------------------
extern "C" void kernel_launch(void* const* d_in, const int* in_sizes, int n_in,
                              void* d_out, int out_size, void* d_ws, size_t ws_size,
                              hipStream_t stream) {
  const float* images = (const float*)d_in[0];   // [B,N,C]
  const int*   edges  = (const int*)d_in[1];     // [B,2,E]
  const float* pseudo = (const float*)d_in[2];   // [B,E,2]
  const float* W1     = (const float*)d_in[3];   // [9,C,O] == [288,32] row-major
  const float* root1  = (const float*)d_in[4];
  const float* b1     = (const float*)d_in[5];
  const float* W2     = (const float*)d_in[6];
  const float* root2  = (const float*)d_in[7];
  const float* b2     = (const float*)d_in[8];
  float* out = (float*)d_out;                    // [B,N,O]
  float* h   = (float*)d_ws;                     // [N,O] scratch, 6.4 MB

  const int nodeBlocks = (NN * NO + 255) / 256;  // 6250
  const int edgeBlocks = NE / 128;               // 6250 exactly (4 waves x 32 edges)

  for (int b = 0; b < NB; ++b) {
    const float* x   = images + (size_t)b * NN * NC;
    const int*   src = edges  + (size_t)b * 2 * NE;
    const int*   dst = src + NE;
    const float* ps  = pseudo + (size_t)b * NE * 2;
    float*       ob  = out + (size_t)b * NN * NO;

    // layer 1 -> h (pre-ReLU; ReLU fused into layer-2 consumers)
    spline_init_kernel<false><<<nodeBlocks, 256, 0, stream>>>(x, root1, b1, h);
    spline_edge_kernel<false><<<edgeBlocks, 128, 0, stream>>>(x, src, dst, ps, W1, h);
    // layer 2 -> out[b]
    spline_init_kernel<true><<<nodeBlocks, 256, 0, stream>>>(h, root2, b2, ob);
    spline_edge_kernel<true><<<edgeBlocks, 128, 0, stream>>>(h, src, dst, ps, W2, ob);
    spline_relu_kernel<<<nodeBlocks, 256, 0, stream>>>(ob);
  }
}